// GaussLayer_10857677324321
// MI455X (gfx1250) — compile-verified
//
#include <hip/hip_runtime.h>
#include <math.h>

typedef __attribute__((ext_vector_type(2))) float v2f;
typedef __attribute__((ext_vector_type(8))) float v8f;

#define GH 256
#define GW 256
#define GC 4
#define GK 4
#define GB 16
#define NPIX (GB * GH * GW)      /* 1,048,576 pixels */
#define NTILES (NPIX / 16)       /* 65,536 WMMA tiles */

// One wave per 16-pixel tile iteration:
//   D(16x16) = A(16x4) * B(4x16) + C(16x16)
//   A: stacked class inverses  row m=(k*4+c) -> (L_k^{-1})[c][:]
//   B: 16 pixels' channel vectors (x)
//   C: b_k[c] = -(L_k^{-1} mu_k)[c]  broadcast over pixels
// => D[m][n] = z-component c of class k at pixel n.
__global__ __launch_bounds__(256) void gauss_wmma_kernel(
    const float* __restrict__ x,
    const float* __restrict__ scale,
    const float* __restrict__ mean,
    float* __restrict__ out)
{
    __shared__ float sA[16][4];   // stacked A = L^{-1} rows
    __shared__ float sQ[16];      // stacked b = -A*mu
    __shared__ float sCst[4];     // per-class: -2*log(2pi) - logdet

    const int tid = threadIdx.x;

    // ---- per-block setup: threads 0..15 each own one (class, component) row
    if (tid < 16) {
        const int k = tid >> 2;
        const int c = tid & 3;
        float L[4][4];
#pragma unroll
        for (int i = 0; i < 4; ++i)
#pragma unroll
            for (int j = 0; j < 4; ++j)
                L[i][j] = (j <= i) ? scale[k * 16 + i * 4 + j] : 0.0f;

        const float d0 = 1.0f / L[0][0];
        const float d1 = 1.0f / L[1][1];
        const float d2 = 1.0f / L[2][2];
        const float d3 = 1.0f / L[3][3];

        float inv[4][4];
#pragma unroll
        for (int i = 0; i < 4; ++i)
#pragma unroll
            for (int j = 0; j < 4; ++j)
                inv[i][j] = 0.0f;
        inv[0][0] = d0;
        inv[1][1] = d1;
        inv[1][0] = -d1 * L[1][0] * d0;
        inv[2][2] = d2;
        inv[2][1] = -d2 * L[2][1] * d1;
        inv[2][0] = -d2 * (L[2][0] * inv[0][0] + L[2][1] * inv[1][0]);
        inv[3][3] = d3;
        inv[3][2] = -d3 * L[3][2] * d2;
        inv[3][1] = -d3 * (L[3][1] * inv[1][1] + L[3][2] * inv[2][1]);
        inv[3][0] = -d3 * (L[3][0] * inv[0][0] + L[3][1] * inv[1][0] + L[3][2] * inv[2][0]);

#pragma unroll
        for (int j = 0; j < 4; ++j)
            sA[tid][j] = inv[c][j];

        float q = 0.0f;
#pragma unroll
        for (int j = 0; j < 4; ++j)
            q = fmaf(inv[c][j], mean[k * 4 + j], q);
        sQ[tid] = -q;

        if (c == 0) {
            const float log2pi = 1.8378770664093453f;
            float ld = __logf(fabsf(L[0][0])) + __logf(fabsf(L[1][1])) +
                       __logf(fabsf(L[2][2])) + __logf(fabsf(L[3][3]));
            sCst[k] = -2.0f * log2pi - ld;   // -0.5*C*log(2pi) - logdet
        }
    }
    __syncthreads();

    // ---- per-lane WMMA constants from LDS
    const int lane = tid & 31;
    const int m    = lane & 15;   // matrix row / pixel-in-tile
    const int hi   = lane >> 4;   // half-wave selector

    v2f amat;
    amat.x = sA[m][2 * hi];       // A cols {0,1} (lanes 0-15) or {2,3} (16-31)
    amat.y = sA[m][2 * hi + 1];

    v8f cmat;
#pragma unroll
    for (int r = 0; r < 8; ++r)
        cmat[r] = sQ[8 * hi + r]; // C rows r (+8 for high half), constant over N

    const float cst0 = sCst[2 * hi];       // class for D rows 0..3
    const float cst1 = sCst[2 * hi + 1];   // class for D rows 4..7

    const size_t outOff0 = (size_t)(2 * hi) * NPIX;
    const size_t outOff1 = outOff0 + NPIX;

    const int waveId = (int)((blockIdx.x * blockDim.x + tid) >> 5);
    const int nWaves = (int)((gridDim.x * blockDim.x) >> 5);

    for (int t = waveId; t < NTILES; t += nWaves) {   // wave-uniform: EXEC stays full
        const int p0 = t * 16;
        // B matrix: channels {2*hi, 2*hi+1} of pixel (p0+m); wave reads 256B contiguous
        const v2f bvec = *(const v2f*)(x + (size_t)(p0 + m) * GC + 2 * hi);

        v8f d = __builtin_amdgcn_wmma_f32_16x16x4_f32(
            /*neg_a=*/false, amat, /*neg_b=*/false, bvec,
            /*c_mod=*/(short)0, cmat, /*reuse_a=*/false, /*reuse_b=*/false);

        const float s0 = d[0] * d[0] + d[1] * d[1] + d[2] * d[2] + d[3] * d[3];
        const float s1 = d[4] * d[4] + d[5] * d[5] + d[6] * d[6] + d[7] * d[7];

        const float pr0 = __expf(fmaf(-0.5f, s0, cst0));
        const float pr1 = __expf(fmaf(-0.5f, s1, cst1));

        out[outOff0 + (size_t)(p0 + m)] = pr0;
        out[outOff1 + (size_t)(p0 + m)] = pr1;
    }
}

extern "C" void kernel_launch(void* const* d_in, const int* in_sizes, int n_in,
                              void* d_out, int out_size, void* d_ws, size_t ws_size,
                              hipStream_t stream) {
    const float* x     = (const float*)d_in[0];  // [B,H,W,C]
    const float* scale = (const float*)d_in[1];  // [K,1,C,C]
    const float* mean  = (const float*)d_in[2];  // [K,1,C]
    float* out = (float*)d_out;                  // [K*B*HW] flat

    (void)in_sizes; (void)n_in; (void)out_size; (void)d_ws; (void)ws_size;

    const dim3 block(256);
    const dim3 grid(2048);   // 16384 wave32s -> 4 tiles each over 65536 tiles
    gauss_wmma_kernel<<<grid, block, 0, stream>>>(x, scale, mean, out);
}